// GATNet_32933809226508
// MI455X (gfx1250) — compile-verified
//
#include <hip/hip_runtime.h>
#include <math.h>

// ---------------------------------------------------------------------------
// 2-layer GAT (PyG GATConv semantics) for MI455X / gfx1250.
// GEMMs use exact-fp32 WMMA: V_WMMA_F32_16X16X4_F32.
// Segment softmax over edges via 3 atomic passes (max / sum / aggregate).
// ---------------------------------------------------------------------------

typedef __attribute__((ext_vector_type(2))) float v2f;
typedef __attribute__((ext_vector_type(8))) float v8f;

#define NEG_SLOPE 0.2f
// enc(-inf): -inf bits = 0xFF800000, negative -> ~bits = 0x007FFFFF
#define ENC_NEG_INF 0x007FFFFFu

__device__ __forceinline__ unsigned enc_f32(float f) {
  unsigned u = __float_as_uint(f);
  return (u & 0x80000000u) ? ~u : (u | 0x80000000u);   // order-preserving map
}
__device__ __forceinline__ float dec_f32(unsigned u) {
  return (u & 0x80000000u) ? __uint_as_float(u & 0x7fffffffu)
                           : __uint_as_float(~u);
}

// --------------------------- generic transpose -----------------------------
__global__ void transpose_k(const float* __restrict__ src, float* __restrict__ dst,
                            int R, int C) {
  int idx = blockIdx.x * blockDim.x + threadIdx.x;
  if (idx >= R * C) return;
  int r = idx / C, c = idx - r * C;
  dst[(size_t)c * R + r] = src[idx];
}

// --------------------------- GEMM1: x[N,512] @ W1[512,64] ------------------
// 8 waves / block. Waves 0-3: rows [row0, row0+16), cols {0,16,32,48}.
// Waves 4-7: rows [row0+16, row0+32). One 16x16 f32 tile per wave.
// WT is W1 transposed: WT[n*512 + k].
__global__ __launch_bounds__(256) void gemm1_wmma(const float* __restrict__ x,
                                                  const float* __restrict__ WT,
                                                  float* __restrict__ H1, int N) {
  const int lane = threadIdx.x & 31;
  const int wave = threadIdx.x >> 5;
  const int half = lane >> 4;   // 0 or 1
  const int l16  = lane & 15;
  const int rowTile = blockIdx.x * 2 + (wave >> 2);
  const int col0 = (wave & 3) * 16;
  const int row0 = rowTile * 16;
  if (row0 >= N) return;  // uniform per wave -> EXEC stays all-ones for WMMA

  int arow = row0 + l16; if (arow >= N) arow = N - 1;   // clamp (dup rows ok)
  const float* aptr = x  + (size_t)arow * 512 + 2 * half;           // A[r][2h+{0,1}]
  const float* bptr = WT + (size_t)(col0 + l16) * 512 + 2 * half;   // B[2h+{0,1}][n]

  v8f acc = {};
#pragma unroll 8
  for (int k = 0; k < 512; k += 4) {
    v2f a = *(const v2f*)(aptr + k);
    v2f b = *(const v2f*)(bptr + k);
    acc = __builtin_amdgcn_wmma_f32_16x16x4_f32(
        /*neg_a=*/false, a, /*neg_b=*/false, b,
        /*c_mod=*/(short)0, acc, /*reuse_a=*/false, /*reuse_b=*/false);
  }
  // D layout: lane holds column n=l16, rows m = v + 8*half (v = 0..7)
  float* outp = H1 + (size_t)(row0 + 8 * half) * 64 + col0 + l16;
  if (row0 + 16 <= N) {           // wave-uniform: full tile, unguarded stores
#pragma unroll
    for (int v = 0; v < 8; ++v) outp[(size_t)v * 64] = acc[v];
  } else {                        // cold partial-tile fallback
#pragma unroll
    for (int v = 0; v < 8; ++v) {
      int r = row0 + 8 * half + v;
      if (r < N) outp[(size_t)v * 64] = acc[v];
    }
  }
}

// --------------------------- GEMM2: h[N,64] @ W2[64,16] --------------------
// One 16x16 tile per wave (cols 0..15), K = 64.
__global__ __launch_bounds__(256) void gemm2_wmma(const float* __restrict__ h,
                                                  const float* __restrict__ WT2,
                                                  float* __restrict__ T2, int N) {
  const int lane = threadIdx.x & 31;
  const int wave = threadIdx.x >> 5;
  const int half = lane >> 4;
  const int l16  = lane & 15;
  const int row0 = (blockIdx.x * 8 + wave) * 16;
  if (row0 >= N) return;

  int arow = row0 + l16; if (arow >= N) arow = N - 1;
  const float* aptr = h   + (size_t)arow * 64 + 2 * half;
  const float* bptr = WT2 + (size_t)l16 * 64 + 2 * half;

  v8f acc = {};
#pragma unroll
  for (int k = 0; k < 64; k += 4) {
    v2f a = *(const v2f*)(aptr + k);
    v2f b = *(const v2f*)(bptr + k);
    acc = __builtin_amdgcn_wmma_f32_16x16x4_f32(false, a, false, b,
                                                (short)0, acc, false, false);
  }
  float* outp = T2 + (size_t)(row0 + 8 * half) * 16 + l16;
  if (row0 + 16 <= N) {
#pragma unroll
    for (int v = 0; v < 8; ++v) outp[(size_t)v * 16] = acc[v];
  } else {
#pragma unroll
    for (int v = 0; v < 8; ++v) {
      int r = row0 + 8 * half + v;
      if (r < N) outp[(size_t)v * 16] = acc[v];
    }
  }
}

// ----------------- per-node attention logits, layer 1 (H=8, C=8) -----------
__global__ void node_alpha1(const float* __restrict__ H1,
                            const float* __restrict__ att_src,
                            const float* __restrict__ att_dst,
                            float* __restrict__ asrc, float* __restrict__ adst, int N) {
  int n = blockIdx.x * blockDim.x + threadIdx.x;
  if (n >= N) return;
  const float* row = H1 + (size_t)n * 64;
#pragma unroll
  for (int h = 0; h < 8; ++h) {
    float s = 0.f, d = 0.f;
#pragma unroll
    for (int c = 0; c < 8; ++c) {
      float v = row[h * 8 + c];
      s += v * att_src[h * 8 + c];
      d += v * att_dst[h * 8 + c];
    }
    asrc[(size_t)n * 8 + h] = s;
    adst[(size_t)n * 8 + h] = d;
  }
}

// ----------------- per-node attention logits, layer 2 (H=1, C=16) ----------
__global__ void node_alpha2(const float* __restrict__ T2,
                            const float* __restrict__ att_src,
                            const float* __restrict__ att_dst,
                            float* __restrict__ asrc, float* __restrict__ adst, int N) {
  int n = blockIdx.x * blockDim.x + threadIdx.x;
  if (n >= N) return;
  const float* row = T2 + (size_t)n * 16;
  float s = 0.f, d = 0.f;
#pragma unroll
  for (int c = 0; c < 16; ++c) {
    float v = row[c];
    s += v * att_src[c];
    d += v * att_dst[c];
  }
  asrc[n] = s;
  adst[n] = d;
}

// --------------------------- init buffers ----------------------------------
__global__ void init_l1(float* __restrict__ out1, unsigned* __restrict__ m1,
                        float* __restrict__ d1, const float* __restrict__ bias, int N) {
  int idx = blockIdx.x * blockDim.x + threadIdx.x;
  if (idx >= N * 64) return;
  out1[idx] = bias[idx & 63];
  if (idx < N * 8) { m1[idx] = ENC_NEG_INF; d1[idx] = 0.f; }
}
__global__ void init_l2(float* __restrict__ o2, unsigned* __restrict__ m2,
                        float* __restrict__ d2, const float* __restrict__ bias, int N) {
  int idx = blockIdx.x * blockDim.x + threadIdx.x;
  if (idx >= N * 16) return;
  o2[idx] = bias[idx & 15];
  if (idx < N) { m2[idx] = ENC_NEG_INF; d2[idx] = 0.f; }
}

// --------------------------- edge passes, layer 1 --------------------------
__device__ __forceinline__ void edge_sd(const int* __restrict__ ei, int e, int E,
                                        int& s, int& d) {
  if (e < E) { s = ei[e]; d = ei[E + e]; }   // edge_index[0,:]=src, [1,:]=dst
  else       { s = d = e - E; }              // self loop
}

__global__ void edge_max1(const int* __restrict__ ei, int E, int N,
                          const float* __restrict__ asrc, const float* __restrict__ adst,
                          unsigned* __restrict__ m1) {
  int e = blockIdx.x * blockDim.x + threadIdx.x;
  if (e >= E + N) return;
  int s, d; edge_sd(ei, e, E, s, d);
#pragma unroll
  for (int h = 0; h < 8; ++h) {
    float a = asrc[(size_t)s * 8 + h] + adst[(size_t)d * 8 + h];
    a = (a > 0.f) ? a : NEG_SLOPE * a;
    atomicMax(&m1[(size_t)d * 8 + h], enc_f32(a));
  }
}

__global__ void edge_sum1(const int* __restrict__ ei, int E, int N,
                          const float* __restrict__ asrc, const float* __restrict__ adst,
                          const unsigned* __restrict__ m1, float* __restrict__ d1) {
  int e = blockIdx.x * blockDim.x + threadIdx.x;
  if (e >= E + N) return;
  int s, d; edge_sd(ei, e, E, s, d);
#pragma unroll
  for (int h = 0; h < 8; ++h) {
    float a = asrc[(size_t)s * 8 + h] + adst[(size_t)d * 8 + h];
    a = (a > 0.f) ? a : NEG_SLOPE * a;
    float ex = expf(a - dec_f32(m1[(size_t)d * 8 + h]));
    atomicAdd(&d1[(size_t)d * 8 + h], ex);
  }
}

__global__ void edge_agg1(const int* __restrict__ ei, int E, int N,
                          const float* __restrict__ asrc, const float* __restrict__ adst,
                          const unsigned* __restrict__ m1, const float* __restrict__ d1,
                          const float* __restrict__ H1, float* __restrict__ out1) {
  int e = blockIdx.x * blockDim.x + threadIdx.x;
  if (e >= E + N) return;
  int s, d; edge_sd(ei, e, E, s, d);
  const float* xs = H1 + (size_t)s * 64;
  float* od = out1 + (size_t)d * 64;
#pragma unroll
  for (int h = 0; h < 8; ++h) {
    float a = asrc[(size_t)s * 8 + h] + adst[(size_t)d * 8 + h];
    a = (a > 0.f) ? a : NEG_SLOPE * a;
    float w = expf(a - dec_f32(m1[(size_t)d * 8 + h])) /
              (d1[(size_t)d * 8 + h] + 1e-16f);
#pragma unroll
    for (int c = 0; c < 8; ++c)
      atomicAdd(&od[h * 8 + c], xs[h * 8 + c] * w);
  }
}

// --------------------------- edge passes, layer 2 --------------------------
__global__ void edge_max2(const int* __restrict__ ei, int E, int N,
                          const float* __restrict__ asrc, const float* __restrict__ adst,
                          unsigned* __restrict__ m2) {
  int e = blockIdx.x * blockDim.x + threadIdx.x;
  if (e >= E + N) return;
  int s, d; edge_sd(ei, e, E, s, d);
  float a = asrc[s] + adst[d];
  a = (a > 0.f) ? a : NEG_SLOPE * a;
  atomicMax(&m2[d], enc_f32(a));
}

__global__ void edge_sum2(const int* __restrict__ ei, int E, int N,
                          const float* __restrict__ asrc, const float* __restrict__ adst,
                          const unsigned* __restrict__ m2, float* __restrict__ d2) {
  int e = blockIdx.x * blockDim.x + threadIdx.x;
  if (e >= E + N) return;
  int s, d; edge_sd(ei, e, E, s, d);
  float a = asrc[s] + adst[d];
  a = (a > 0.f) ? a : NEG_SLOPE * a;
  atomicAdd(&d2[d], expf(a - dec_f32(m2[d])));
}

__global__ void edge_agg2(const int* __restrict__ ei, int E, int N,
                          const float* __restrict__ asrc, const float* __restrict__ adst,
                          const unsigned* __restrict__ m2, const float* __restrict__ d2,
                          const float* __restrict__ T2, float* __restrict__ o2) {
  int e = blockIdx.x * blockDim.x + threadIdx.x;
  if (e >= E + N) return;
  int s, d; edge_sd(ei, e, E, s, d);
  float a = asrc[s] + adst[d];
  a = (a > 0.f) ? a : NEG_SLOPE * a;
  float w = expf(a - dec_f32(m2[d])) / (d2[d] + 1e-16f);
  const float* xs = T2 + (size_t)s * 16;
  float* od = o2 + (size_t)d * 16;
#pragma unroll
  for (int c = 0; c < 16; ++c)
    atomicAdd(&od[c], xs[c] * w);
}

// --------------------------- ELU + log_softmax -----------------------------
__global__ void final_k(const float* __restrict__ o2, float* __restrict__ out, int N) {
  int n = blockIdx.x * blockDim.x + threadIdx.x;
  if (n >= N) return;
  const float* r = o2 + (size_t)n * 16;
  float v[16];
  float mx = -INFINITY;
#pragma unroll
  for (int i = 0; i < 16; ++i) {
    float t = r[i];
    t = (t > 0.f) ? t : (expf(t) - 1.f);  // ELU(alpha=1)
    v[i] = t;
    mx = fmaxf(mx, t);
  }
  float s = 0.f;
#pragma unroll
  for (int i = 0; i < 16; ++i) s += expf(v[i] - mx);
  float lse = mx + logf(s);
  float* w = out + (size_t)n * 16;
#pragma unroll
  for (int i = 0; i < 16; ++i) w[i] = v[i] - lse;
}

// ---------------------------------------------------------------------------
extern "C" void kernel_launch(void* const* d_in, const int* in_sizes, int n_in,
                              void* d_out, int out_size, void* d_ws, size_t ws_size,
                              hipStream_t stream) {
  const float* x   = (const float*)d_in[0];
  const int*   ei  = (const int*)d_in[1];
  const float* W1  = (const float*)d_in[2];
  const float* at_s1 = (const float*)d_in[3];
  const float* at_d1 = (const float*)d_in[4];
  const float* b1  = (const float*)d_in[5];
  const float* W2  = (const float*)d_in[6];
  const float* at_s2 = (const float*)d_in[7];
  const float* at_d2 = (const float*)d_in[8];
  const float* b2  = (const float*)d_in[9];

  const int N = in_sizes[0] / 512;
  const int E = in_sizes[1] / 2;
  const int ET = E + N;                // edges + self loops
  float* out = (float*)d_out;

  // ---- workspace carve-up (floats) ----
  float* ws = (float*)d_ws;
  size_t off = 0;
  float*    WT1   = ws + off; off += (size_t)64 * 512;
  float*    WT2   = ws + off; off += (size_t)16 * 64;
  float*    H1    = ws + off; off += (size_t)N * 64;   // x @ W1
  float*    asrc1 = ws + off; off += (size_t)N * 8;
  float*    adst1 = ws + off; off += (size_t)N * 8;
  unsigned* m1    = (unsigned*)(ws + off); off += (size_t)N * 8;
  float*    d1    = ws + off; off += (size_t)N * 8;
  float*    out1  = ws + off; off += (size_t)N * 64;   // layer-1 output h
  float*    T2    = ws + off; off += (size_t)N * 16;   // h @ W2
  float*    asrc2 = ws + off; off += (size_t)N;
  float*    adst2 = ws + off; off += (size_t)N;
  unsigned* m2    = (unsigned*)(ws + off); off += (size_t)N;
  float*    d2    = ws + off; off += (size_t)N;
  float*    o2    = ws + off; off += (size_t)N * 16;   // layer-2 aggregate

  const int TB = 256;
  // weight transposes (so WMMA B fragments are contiguous float2 loads)
  transpose_k<<<(512 * 64 + TB - 1) / TB, TB, 0, stream>>>(W1, WT1, 512, 64);
  transpose_k<<<(64 * 16 + TB - 1) / TB, TB, 0, stream>>>(W2, WT2, 64, 16);

  // layer 1
  gemm1_wmma<<<(N + 31) / 32, TB, 0, stream>>>(x, WT1, H1, N);
  node_alpha1<<<(N + TB - 1) / TB, TB, 0, stream>>>(H1, at_s1, at_d1, asrc1, adst1, N);
  init_l1<<<((size_t)N * 64 + TB - 1) / TB, TB, 0, stream>>>(out1, m1, d1, b1, N);
  edge_max1<<<(ET + TB - 1) / TB, TB, 0, stream>>>(ei, E, N, asrc1, adst1, m1);
  edge_sum1<<<(ET + TB - 1) / TB, TB, 0, stream>>>(ei, E, N, asrc1, adst1, m1, d1);
  edge_agg1<<<(ET + TB - 1) / TB, TB, 0, stream>>>(ei, E, N, asrc1, adst1, m1, d1, H1, out1);

  // layer 2
  gemm2_wmma<<<(N + 127) / 128, TB, 0, stream>>>(out1, WT2, T2, N);
  node_alpha2<<<(N + TB - 1) / TB, TB, 0, stream>>>(T2, at_s2, at_d2, asrc2, adst2, N);
  init_l2<<<((size_t)N * 16 + TB - 1) / TB, TB, 0, stream>>>(o2, m2, d2, b2, N);
  edge_max2<<<(ET + TB - 1) / TB, TB, 0, stream>>>(ei, E, N, asrc2, adst2, m2);
  edge_sum2<<<(ET + TB - 1) / TB, TB, 0, stream>>>(ei, E, N, asrc2, adst2, m2, d2);
  edge_agg2<<<(ET + TB - 1) / TB, TB, 0, stream>>>(ei, E, N, asrc2, adst2, m2, d2, T2, o2);

  // ELU + log_softmax
  final_k<<<(N + TB - 1) / TB, TB, 0, stream>>>(o2, out, N);
}